// NNNet_60086592471048
// MI455X (gfx1250) — compile-verified
//
#include <hip/hip_runtime.h>

typedef __attribute__((ext_vector_type(2))) float v2f;
typedef __attribute__((ext_vector_type(8))) float v8f;

#define NS   1024
#define NQn  8192
#define BB   16
#define NDV  64

// Block: 256 threads = 8 wave32. Each wave owns a 16-query tile; block covers
// 128 queries of one batch. gridDim = (NQ/128, B).
//
// WMMA mapping (A = samples, B = queries):
//   A row (sample m):  [sx, sy, s2, 1 ]   (16x4 f32 fragment, from LDS)
//   B col (query  n):  [-2qx, -2qy, 1, q2] (4x16 f32 fragment, in regs)
//   C(m,n) = s2 + q2 - 2*(s.q) = exact squared distance.
// C layout: lane holds 8 samples (M = r + 8*half) of ONE query (N = lane&15),
// so the in-loop argmin collapses to a v_min3 tree + tile tracking (7 VALU).
__global__ __launch_bounds__(256) void nn1_wmma_kernel(
    const float* __restrict__ sample_vals,   // [NS, B, 64]
    const float* __restrict__ sample_posns,  // [NS, B, 2]
    const float* __restrict__ query_posns,   // [NQ, B, 2]
    float* __restrict__ out)                 // [NQ, B, 64]
{
    // Per sample s two v2f entries (WMMA-A layout):
    //   s_tab[2s]   = { sx, sy }            (K0,K1 for lanes 0-15)
    //   s_tab[2s+1] = { sx^2+sy^2, 1.0 }    (K2,K3 for lanes 16-31)
    __shared__ v2f s_tab[NS * 2];
    __shared__ int s_bestIdx[128];

    const int b    = blockIdx.y;
    const int tid  = threadIdx.x;
    const int lane = tid & 31;
    const int wave = tid >> 5;
    const int col  = lane & 15;   // query column N / sample sub-row selector
    const int half = lane >> 4;   // 0: K0-K1 lanes, 1: K2-K3 lanes

    // ---- Build sample position table in LDS (1024 samples, 16 KB) ----
    const v2f* sp2 = (const v2f*)sample_posns;   // [NS*B] of (x,y)
    for (int s = tid; s < NS; s += 256) {
        v2f xy = sp2[s * BB + b];
        v2f lo, hi;
        lo.x = xy.x;                     lo.y = xy.y;
        hi.x = xy.x * xy.x + xy.y * xy.y; hi.y = 1.0f;
        s_tab[2 * s]     = lo;
        s_tab[2 * s + 1] = hi;
    }
    __syncthreads();

    // ---- B fragment: this wave's 16 queries ----
    const int qbase = blockIdx.x * 128 + wave * 16;
    const int q     = qbase + col;
    v2f qxy = ((const v2f*)query_posns)[q * BB + b];
    const float q2 = qxy.x * qxy.x + qxy.y * qxy.y;
    v2f bq;
    bq.x = half ? 1.0f : -2.0f * qxy.x;  // lanes 0-15: (-2qx,-2qy); 16-31: (1,q2)
    bq.y = half ? q2   : -2.0f * qxy.y;

    // ---- Scan 64 sample tiles: min3-tree + winning-tile tracking ----
    float best     = 3.402823e38f;
    int   bestTile = 0;

    for (int t = 0; t < NS / 16; ++t) {
        v2f af = s_tab[2 * (t * 16 + col) + half];   // A fragment (ds_load_b64)

        v8f c = {};
        c = __builtin_amdgcn_wmma_f32_16x16x4_f32(
                /*neg_a=*/false, af, /*neg_b=*/false, bq,
                /*c_mod=*/(short)0, c, /*reuse_a=*/false, /*reuse_b=*/false);

        // min of 8 distances for this lane's query: 2x v_min3 + min + v_min3
        float ma = fminf(fminf(c[0], c[1]), c[2]);
        float mb = fminf(fminf(c[3], c[4]), c[5]);
        float mc = fminf(ma, mb);
        float m  = fminf(fminf(c[6], c[7]), mc);

        const bool lt = m < best;            // strict: keep earliest tile
        bestTile = lt ? t : bestTile;
        best     = lt ? m : best;
    }

    // ---- Combine the two lane-halves (samples 0-7 vs 8-15 of each tile) ----
    {
        const int paddr = (lane ^ 16) << 2;
        float od = __int_as_float(
            __builtin_amdgcn_ds_bpermute(paddr, __float_as_int(best)));
        int   ot = __builtin_amdgcn_ds_bpermute(paddr, bestTile);
        // prefer lower sample index (half 0) on exact ties
        const bool take = half ? !(best < od) : (od < best);
        const int candHalf = take ? (half ^ 1) : half;
        bestTile = take ? ot : bestTile;

        // ---- Index recovery: rescore the 8 winning candidates from LDS ----
        const int base = bestTile * 16 + candHalf * 8;
        float bd = 3.402823e38f;
        int   bi = base;
        #pragma unroll
        for (int r = 0; r < 8; ++r) {
            v2f sxy = s_tab[2 * (base + r)];         // {sx, sy}
            v2f shi = s_tab[2 * (base + r) + 1];     // {s2, 1}
            float qs = qxy.x * sxy.x + qxy.y * sxy.y;
            float d  = (q2 - 2.0f * qs) + shi.x;     // mirror reference formula
            const bool l2 = d < bd;                  // strict: lowest index wins
            bi = l2 ? (base + r) : bi;
            bd = l2 ? d : bd;
        }
        if (half == 0) s_bestIdx[wave * 16 + col] = bi;
    }
    __syncthreads();

    // ---- Gather + coalesced output: 16 lanes move one 256B value row ----
    for (int p = 0; p < 8; ++p) {
        const int row  = p * 2 + (lane >> 4);          // 0..15 within wave tile
        const int qg   = qbase + row;
        const int sidx = s_bestIdx[wave * 16 + row];
        const float4* src =
            (const float4*)(sample_vals + (size_t)(sidx * BB + b) * NDV) + (lane & 15);
        float4* dst =
            (float4*)(out + (size_t)(qg * BB + b) * NDV) + (lane & 15);
        *dst = *src;
    }
}

extern "C" void kernel_launch(void* const* d_in, const int* in_sizes, int n_in,
                              void* d_out, int out_size, void* d_ws, size_t ws_size,
                              hipStream_t stream) {
    const float* sample_vals  = (const float*)d_in[0];  // [1024,16,64]
    const float* sample_posns = (const float*)d_in[1];  // [1024,16,2]
    const float* query_posns  = (const float*)d_in[2];  // [8192,16,2]
    float* out = (float*)d_out;                         // [8192,16,64]

    dim3 grid(NQn / 128, BB);   // (64, 16)
    nn1_wmma_kernel<<<grid, 256, 0, stream>>>(sample_vals, sample_posns,
                                              query_posns, out);
}